// Net_45268955300358
// MI455X (gfx1250) — compile-verified
//
#include <hip/hip_runtime.h>
#include <hip/hip_bf16.h>

typedef __attribute__((ext_vector_type(2))) float v2f;
typedef __attribute__((ext_vector_type(8))) float v8f;

__device__ __forceinline__ v8f wmma_f32(v2f a, v2f b, v8f c) {
  // D = A(16x4 f32) * B(4x16 f32) + C(16x16 f32)
  return __builtin_amdgcn_wmma_f32_16x16x4_f32(false, a, false, b, (short)0, c, false, false);
}

// ---------------------------------------------------------------- utilities
__global__ void k_fill(float* __restrict__ p, float v, int n) {
  int i = blockIdx.x * 256 + threadIdx.x;
  if (i < n) p[i] = v;
}

__global__ void k_degree(const int* __restrict__ dst, float* __restrict__ deg, int E) {
  int e = blockIdx.x * 256 + threadIdx.x;
  if (e < E) unsafeAtomicAdd(&deg[dst[e]], 1.0f);
}

__global__ void k_rsqrt_inplace(float* __restrict__ p, int n) {
  int i = blockIdx.x * 256 + threadIdx.x;
  if (i < n) p[i] = rsqrtf(p[i]);
}

__global__ void k_relu(float* __restrict__ p, long long n) {
  long long i = (long long)blockIdx.x * 256 + threadIdx.x;
  if (i < n) p[i] = fmaxf(p[i], 0.0f);
}

// ------------------------------------------------------- generic WMMA GEMM
// C[M,FOUT] = A[M,K] @ W[K,FOUT]; one wave computes one 16x16 tile of C.
template<int K, int FOUT, int NTILES>
__global__ void k_gemm_wmma(const float* __restrict__ A, const float* __restrict__ W,
                            float* __restrict__ C, int M)
{
  const int lane = threadIdx.x & 31;
  const int wave = threadIdx.x >> 5;
  const int tid  = blockIdx.x * (blockDim.x >> 5) + wave;
  const int mt = tid / NTILES;            // constexpr pow2 -> shift
  const int nt = tid - mt * NTILES;
  if (mt * 16 >= M) return;               // wave-uniform: EXEC stays all-1s for WMMA
  const int mrow = mt * 16 + (lane & 15);
  const int ncol = nt * 16 + (lane & 15);
  const int koff = (lane >> 4) * 2;
  const float* Ap = A + (size_t)mrow * K;
  v8f c = {};
  #pragma unroll
  for (int k0 = 0; k0 < K; k0 += 4) {
    v2f a, b;
    a.x = Ap[k0 + koff];
    a.y = Ap[k0 + koff + 1];
    b.x = W[(size_t)(k0 + koff)     * FOUT + ncol];
    b.y = W[(size_t)(k0 + koff + 1) * FOUT + ncol];
    c = wmma_f32(a, b, c);
  }
  const int rbase = mt * 16 + 8 * (lane >> 4);
  #pragma unroll
  for (int r = 0; r < 8; ++r)
    C[(size_t)(rbase + r) * FOUT + ncol] = c[r];
}

// ------------------------------------------- GCN aggregation (edge scatter)
// Pre-pass: agg[i,f] = dinv[i]^2 * h[i,f] + bias[f]   (self-loop + bias)
template<int F>
__global__ void k_self_bias(const float* __restrict__ h, const float* __restrict__ dinv,
                            const float* __restrict__ bias, float* __restrict__ agg, int N)
{
  long long t = (long long)blockIdx.x * blockDim.x + threadIdx.x;
  int i = (int)(t / F);
  int f = (int)(t - (long long)i * F);
  if (i >= N) return;
  float di = dinv[i];
  size_t idx = (size_t)i * F + f;
  agg[idx] = di * di * h[idx] + bias[f];
}

// Edge pass: agg[dst] += dinv[src]*dinv[dst] * h[src]. float4 gathers, f32 atomics.
template<int F>
__global__ void k_aggregate(const int* __restrict__ src, const int* __restrict__ dst,
                            const float* __restrict__ dinv, const float* __restrict__ h,
                            float* __restrict__ agg, int E)
{
  constexpr int LPE = F / 4;  // lanes per edge
  long long t = (long long)blockIdx.x * blockDim.x + threadIdx.x;
  int e  = (int)(t / LPE);
  int f4 = (int)(t - (long long)e * LPE) * 4;
  if (e >= E) return;
  int s = src[e], d = dst[e];
  float norm = dinv[s] * dinv[d];
  const float4 hv = *(const float4*)(h + (size_t)s * F + f4);
  float* ap = agg + (size_t)d * F + f4;
  unsafeAtomicAdd(ap + 0, norm * hv.x);
  unsafeAtomicAdd(ap + 1, norm * hv.y);
  unsafeAtomicAdd(ap + 2, norm * hv.z);
  unsafeAtomicAdd(ap + 3, norm * hv.w);
}

// ------------------- fused head: ELU(agg2@Wf1+bf1) @ Wf2 + bf2, log_softmax
// One wave per 16-node tile. FC1 result lives in padded LDS.
__global__ void k_head(const float* __restrict__ agg2, const float* __restrict__ Wf1,
                       const float* __restrict__ bf1, const float* __restrict__ Wf2,
                       const float* __restrict__ bf2, float* __restrict__ out, int N)
{
  constexpr int TS = 132;                 // padded stride: kills DS bank conflicts
  __shared__ float t_lds[16 * TS];        // 16x128 ELU(FC1) tile
  __shared__ float l_lds[16 * 17];        // 16x16 logits tile (10 valid cols)
  const int lane  = threadIdx.x & 31;
  const int mt    = blockIdx.x;
  const int mrow  = mt * 16 + (lane & 15);
  const int nn    = lane & 15;
  const int koff  = (lane >> 4) * 2;
  const int rhalf = 8 * (lane >> 4);

  // FC1: t = ELU(agg2[16x64] @ Wf1[64x128] + bf1)
  const float* Ap = agg2 + (size_t)mrow * 64;
  for (int ntile = 0; ntile < 8; ++ntile) {
    v8f c = {};
    #pragma unroll
    for (int k0 = 0; k0 < 64; k0 += 4) {
      v2f a, b;
      a.x = Ap[k0 + koff];
      a.y = Ap[k0 + koff + 1];
      b.x = Wf1[(size_t)(k0 + koff)     * 128 + ntile * 16 + nn];
      b.y = Wf1[(size_t)(k0 + koff + 1) * 128 + ntile * 16 + nn];
      c = wmma_f32(a, b, c);
    }
    #pragma unroll
    for (int r = 0; r < 8; ++r) {
      float v = c[r] + bf1[ntile * 16 + nn];
      v = (v > 0.0f) ? v : expm1f(v);     // ELU
      t_lds[(r + rhalf) * TS + ntile * 16 + nn] = v;
    }
  }
  __syncthreads();

  // FC2: logits = t[16x128] @ Wf2[128x10, N-padded to 16] + bf2
  v8f c2 = {};
  #pragma unroll
  for (int k0 = 0; k0 < 128; k0 += 4) {
    v2f a, b;
    a.x = t_lds[(lane & 15) * TS + k0 + koff];
    a.y = t_lds[(lane & 15) * TS + k0 + koff + 1];
    b.x = (nn < 10) ? Wf2[(size_t)(k0 + koff)     * 10 + nn] : 0.0f;
    b.y = (nn < 10) ? Wf2[(size_t)(k0 + koff + 1) * 10 + nn] : 0.0f;
    c2 = wmma_f32(a, b, c2);
  }
  #pragma unroll
  for (int r = 0; r < 8; ++r)
    l_lds[(r + rhalf) * 17 + nn] = c2[r] + ((nn < 10) ? bf2[nn] : 0.0f);
  __syncthreads();

  // log_softmax over the 10 classes; one lane per node row
  if (lane < 16) {
    int node = mt * 16 + lane;
    if (node < N) {
      float mx = -3.402823466e38f;
      #pragma unroll
      for (int j = 0; j < 10; ++j) mx = fmaxf(mx, l_lds[lane * 17 + j]);
      float s = 0.0f;
      #pragma unroll
      for (int j = 0; j < 10; ++j) s += expf(l_lds[lane * 17 + j] - mx);
      float lse = mx + logf(s);
      #pragma unroll
      for (int j = 0; j < 10; ++j)
        out[(size_t)node * 10 + j] = l_lds[lane * 17 + j] - lse;
    }
  }
}

// ---------------------------------------------------------------- launcher
extern "C" void kernel_launch(void* const* d_in, const int* in_sizes, int n_in,
                              void* d_out, int out_size, void* d_ws, size_t ws_size,
                              hipStream_t stream)
{
  const float* x   = (const float*)d_in[0];
  const int*   ei  = (const int*)  d_in[1];
  const float* W1  = (const float*)d_in[2];
  const float* b1  = (const float*)d_in[3];
  const float* W2  = (const float*)d_in[4];
  const float* b2  = (const float*)d_in[5];
  const float* Wf1 = (const float*)d_in[6];
  const float* bf1 = (const float*)d_in[7];
  const float* Wf2 = (const float*)d_in[8];
  const float* bf2 = (const float*)d_in[9];
  float* out = (float*)d_out;

  const int N = in_sizes[0] / 128;
  const int E = in_sizes[1] / 2;
  const int* src = ei;
  const int* dst = ei + E;

  // workspace: dinv[N] | h1[N*32] | agg1[N*32] | h2[N*64] | agg2[N*64]  (~77 MB)
  float* ws   = (float*)d_ws;
  float* dinv = ws;
  float* h1   = dinv + N;
  float* agg1 = h1   + (size_t)N * 32;
  float* h2   = agg1 + (size_t)N * 32;
  float* agg2 = h2   + (size_t)N * 64;

  auto cdiv = [](long long a, long long b) { return (int)((a + b - 1) / b); };
  const int mtiles = cdiv(N, 16);

  // degrees -> dinv = rsqrt(deg + 1)
  k_fill<<<cdiv(N, 256), 256, 0, stream>>>(dinv, 1.0f, N);
  k_degree<<<cdiv(E, 256), 256, 0, stream>>>(dst, dinv, E);
  k_rsqrt_inplace<<<cdiv(N, 256), 256, 0, stream>>>(dinv, N);

  // ---- GCN layer 1: h1 = x @ W1; agg1 = scatter + self + b1; ReLU
  k_gemm_wmma<128, 32, 2><<<cdiv((long long)mtiles * 2, 4), 128, 0, stream>>>(x, W1, h1, N);
  k_self_bias<32><<<cdiv((long long)N * 32, 256), 256, 0, stream>>>(h1, dinv, b1, agg1, N);
  k_aggregate<32><<<cdiv((long long)E * 8, 256), 256, 0, stream>>>(src, dst, dinv, h1, agg1, E);
  k_relu<<<cdiv((long long)N * 32, 256), 256, 0, stream>>>(agg1, (long long)N * 32);

  // ---- GCN layer 2: h2 = relu(agg1) @ W2; agg2 = scatter + self + b2
  k_gemm_wmma<32, 64, 4><<<cdiv((long long)mtiles * 4, 4), 128, 0, stream>>>(agg1, W2, h2, N);
  k_self_bias<64><<<cdiv((long long)N * 64, 256), 256, 0, stream>>>(h2, dinv, b2, agg2, N);
  k_aggregate<64><<<cdiv((long long)E * 16, 256), 256, 0, stream>>>(src, dst, dinv, h2, agg2, E);

  // ---- fused head: FC1 + ELU + FC2 + log_softmax
  k_head<<<mtiles, 32, 0, stream>>>(agg2, Wf1, bf1, Wf2, bf2, out, N);
}